// myLoss_39926015983879
// MI455X (gfx1250) — compile-verified
//
#include <hip/hip_runtime.h>
#include <hip/hip_bf16.h>
#include <stdint.h>

// ---------------------------------------------------------------------------
// myLoss on MI455X (gfx1250):
//   per-row logsumexp over 32000 f32 logits, ce = lse - x[tag],
//   loss = sum_rows relu(1.4 - abn*ce).
// Memory-bound: 1.05 GB @ 23.3 TB/s => ~45us floor. Strategy: stage each full
// row (125 KB) into LDS (320 KB/WGP) with GLOBAL_LOAD_ASYNC_TO_LDS_B128
// (ASYNCcnt pipeline), then do max + sum-exp passes entirely out of LDS so
// HBM is touched exactly once. No GEMM structure => WMMA not applicable.
// ---------------------------------------------------------------------------

#define THREADS 256
#define NWAVES (THREADS / 32)
#define BND_F 1.4f

typedef int v4i __attribute__((ext_vector_type(4)));
typedef __attribute__((address_space(1))) v4i glob_v4i;   // global int4
typedef __attribute__((address_space(3))) v4i lds_v4i;    // LDS int4

// Async DMA of 16 bytes global -> LDS (CDNA5, ASYNCcnt-tracked). Falls back
// to a synchronous vector copy if this toolchain lacks the builtin.
__device__ __forceinline__ void async_copy_b128(const void* gsrc, void* ldst) {
#if __has_builtin(__builtin_amdgcn_global_load_async_to_lds_b128)
    __builtin_amdgcn_global_load_async_to_lds_b128(
        (glob_v4i*)(uintptr_t)gsrc,
        (lds_v4i*)(uint32_t)(uintptr_t)ldst,  // low 32 bits == LDS offset
        /*imm offset*/ 0, /*cpol*/ 0);
#else
    *(float4*)ldst = *(const float4*)gsrc;
#endif
}

__device__ __forceinline__ void wait_async_zero() {
#if __has_builtin(__builtin_amdgcn_s_wait_asynccnt)
    __builtin_amdgcn_s_wait_asynccnt(0);
#else
    asm volatile("s_wait_asynccnt 0" ::: "memory");
#endif
}

__global__ void __launch_bounds__(THREADS)
row_loss_kernel(const float* __restrict__ logits,
                const int* __restrict__ labels,
                float* __restrict__ row_loss,
                int C) {
    extern __shared__ float lds_row[];       // C floats (dynamic, 125 KB)
    __shared__ float red[NWAVES];            // cross-wave reduction scratch

    const int row = blockIdx.x;
    const int tid = threadIdx.x;
    const float* grow = logits + (size_t)row * (size_t)C;

    const int C4 = C >> 2;                   // full float4 chunks
    const float4* gsrc4 = (const float4*)grow;

    // ---- Stage entire row into LDS via async DMA (ASYNCcnt-tracked) ----
    for (int i = tid; i < C4; i += THREADS) {
        async_copy_b128(gsrc4 + i, lds_row + 4 * i);
    }
    // Scalar tail (C % 4) — synchronous, normally dead (C = 32000).
    for (int i = (C4 << 2) + tid; i < C; i += THREADS) {
        lds_row[i] = grow[i];
    }
    wait_async_zero();       // this wave's DMAs done (LDS writes visible)
    __syncthreads();         // all waves' DMAs done

    // ---- Pass 1: row max (LDS only) ----
    float m = -3.402823466e38f;
    for (int i = tid; i < C4; i += THREADS) {
        float4 v = *(const float4*)(lds_row + 4 * i);
        m = fmaxf(m, fmaxf(fmaxf(v.x, v.y), fmaxf(v.z, v.w)));
    }
    for (int i = (C4 << 2) + tid; i < C; i += THREADS) m = fmaxf(m, lds_row[i]);
#pragma unroll
    for (int off = 16; off > 0; off >>= 1)
        m = fmaxf(m, __shfl_xor(m, off, 32));            // wave32 reduce
    if ((tid & 31) == 0) red[tid >> 5] = m;
    __syncthreads();
    float rowmax = red[0];
#pragma unroll
    for (int w = 1; w < NWAVES; ++w) rowmax = fmaxf(rowmax, red[w]);
    __syncthreads();         // release `red` for reuse

    // ---- Pass 2: sum of exp(x - max) (LDS only) ----
    float s = 0.0f;
    for (int i = tid; i < C4; i += THREADS) {
        float4 v = *(const float4*)(lds_row + 4 * i);
        s += __expf(v.x - rowmax) + __expf(v.y - rowmax) +
             __expf(v.z - rowmax) + __expf(v.w - rowmax);
    }
    for (int i = (C4 << 2) + tid; i < C; i += THREADS)
        s += __expf(lds_row[i] - rowmax);
#pragma unroll
    for (int off = 16; off > 0; off >>= 1)
        s += __shfl_xor(s, off, 32);
    if ((tid & 31) == 0) red[tid >> 5] = s;
    __syncthreads();

    if (tid == 0) {
        float tot = 0.0f;
#pragma unroll
        for (int w = 0; w < NWAVES; ++w) tot += red[w];
        const int   tag = labels[2 * row];               // 0 or 1
        const float abn = (labels[2 * row + 1] == 0) ? -1.0f : 1.0f;
        const float x_tag = lds_row[tag];
        const float ce = (__logf(tot) + rowmax) - x_tag; // -log_softmax[tag]
        const float t = BND_F - abn * ce;
        row_loss[row] = t > 0.0f ? t : 0.0f;
    }
}

// Deterministic fixed-order final sum of B per-row losses -> d_out[0].
__global__ void __launch_bounds__(THREADS)
final_reduce_kernel(const float* __restrict__ row_loss,
                    float* __restrict__ out, int B) {
    __shared__ float red[NWAVES];
    const int tid = threadIdx.x;
    float s = 0.0f;
    for (int i = tid; i < B; i += THREADS) s += row_loss[i];
#pragma unroll
    for (int off = 16; off > 0; off >>= 1)
        s += __shfl_xor(s, off, 32);
    if ((tid & 31) == 0) red[tid >> 5] = s;
    __syncthreads();
    if (tid == 0) {
        float tot = 0.0f;
#pragma unroll
        for (int w = 0; w < NWAVES; ++w) tot += red[w];
        out[0] = tot;
    }
}

extern "C" void kernel_launch(void* const* d_in, const int* in_sizes, int n_in,
                              void* d_out, int out_size, void* d_ws, size_t ws_size,
                              hipStream_t stream) {
    const float* logits = (const float*)d_in[0];   // (B, C) f32
    const int*   labels = (const int*)d_in[1];     // (B, 2) int32
    (void)n_in; (void)out_size; (void)ws_size;

    const int B = in_sizes[1] / 2;                 // 8192
    const int C = in_sizes[0] / B;                 // 32000

    float* row_loss = (float*)d_ws;                // B floats of scratch

    const size_t lds_bytes = (size_t)C * sizeof(float);  // 125 KB / WGP block
    row_loss_kernel<<<B, THREADS, lds_bytes, stream>>>(logits, labels, row_loss, C);
    final_reduce_kernel<<<1, THREADS, 0, stream>>>(row_loss, (float*)d_out, B);
}